// GCN_63848983822675
// MI455X (gfx1250) — compile-verified
//
#include <hip/hip_runtime.h>
#include <math.h>

// Problem constants (fixed by the reference).
#define IN_DIM  100
#define HIDDEN  16
#define NCLS    40

typedef __attribute__((ext_vector_type(2))) float v2f;
typedef __attribute__((ext_vector_type(8))) float v8f;

// ---------------------------------------------------------------------------
// D = A(16x4, f32) * B(4x16, f32) + C(16x16, f32) on one wave32.
// A layout: lane l (l<16): holds A[l][0],A[l][1]; lane l+16: A[l][2],A[l][3].
// B layout: lane l (l<16): holds B[0][l],B[1][l]; lane l+16: B[2][l],B[3][l].
// C/D layout: VGPR g -> row (g + 8*(lane>>4)), col (lane&15).
// ---------------------------------------------------------------------------
static __device__ inline v8f wmma_f32_16x16x4(v2f a, v2f b, v8f c) {
#if __has_builtin(__builtin_amdgcn_wmma_f32_16x16x4_f32)
  return __builtin_amdgcn_wmma_f32_16x16x4_f32(
      /*neg_a=*/false, a, /*neg_b=*/false, b,
      /*c_mod=*/(short)0, c, /*reuse_a=*/false, /*reuse_b=*/false);
#else
  // Shuffle-based emulation (correct but slow) so the file always compiles.
  const int lane = (int)(threadIdx.x & 31u);
  const int nsel = lane & 15;
  v8f d = c;
#pragma unroll
  for (int g = 0; g < 8; ++g) {
    const int m = g + 8 * (lane >> 4);
    float acc = 0.f;
#pragma unroll
    for (int k = 0; k < 4; ++k) {
      const int srcA = m + 16 * (k >> 1);
      const int srcB = nsel + 16 * (k >> 1);
      const float av = __shfl((k & 1) ? a.y : a.x, srcA, 32);
      const float bv = __shfl((k & 1) ? b.y : b.x, srcB, 32);
      acc += av * bv;
    }
    d[g] += acc;
  }
  return d;
#endif
}

// ---------------------------------------------------------------------------
// Zero-fill workspace accumulators (they receive atomic adds each call).
// ---------------------------------------------------------------------------
__global__ void gcn_zero_kernel(float* __restrict__ p, int n) {
  int i = blockIdx.x * blockDim.x + threadIdx.x;
  int stride = gridDim.x * blockDim.x;
  for (; i < n; i += stride) p[i] = 0.0f;
}

// ---------------------------------------------------------------------------
// Y[N,16] = X[N,100] @ W1[100,16]. One wave per 16-row tile; 25 WMMAs (K=100).
// ---------------------------------------------------------------------------
__global__ __launch_bounds__(32) void gcn_gemm1_wmma(
    const float* __restrict__ X, const float* __restrict__ W1,
    float* __restrict__ Y, int N) {
  const int lane = (int)threadIdx.x;           // 0..31
  const int r0   = blockIdx.x * 16;
  const int col  = lane & 15;
  const int kh   = lane >> 4;                  // 0 or 1 (K half)
  const int rowA = min(r0 + col, N - 1);       // A-matrix row for this lane
  const float* __restrict__ Xr = X + (size_t)rowA * IN_DIM;

  v8f acc = {};
#pragma unroll 5
  for (int k0 = 0; k0 < IN_DIM; k0 += 4) {     // 25 steps
    const int ka = k0 + 2 * kh;                // even -> 8B-aligned float2
    v2f a;
    a.x = Xr[ka];
    a.y = Xr[ka + 1];
    v2f b;
    b.x = W1[ka * HIDDEN + col];
    b.y = W1[(ka + 1) * HIDDEN + col];
    acc = wmma_f32_16x16x4(a, b, acc);
  }

  // Epilogue: uniform (scalar) branch selects the unguarded fast path; for
  // N % 16 == 0 (true here: 50000 = 3125*16) every wave takes it, so stores
  // issue as a clean clause with no per-lane exec manipulation.
  float* __restrict__ Yr = Y + (size_t)(r0 + 8 * kh) * HIDDEN + col;
  if (r0 + 16 <= N) {
#pragma unroll
    for (int g = 0; g < 8; ++g) Yr[g * HIDDEN] = acc[g];
  } else {
#pragma unroll
    for (int g = 0; g < 8; ++g) {
      const int row = r0 + g + 8 * kh;
      if (row < N) Y[row * HIDDEN + col] = acc[g];
    }
  }
}

// ---------------------------------------------------------------------------
// Z[dst,:] += val * f(Xin[src,:]) over all edges, 16 lanes per edge.
// Gather and scatter are 64B-contiguous per edge-group. f = identity or ReLU.
// ---------------------------------------------------------------------------
__global__ __launch_bounds__(256) void gcn_spmm_kernel(
    const int* __restrict__ esrc, const int* __restrict__ edst,
    const float* __restrict__ evals, const float* __restrict__ Xin,
    float* __restrict__ Z, int E, int doRelu) {
  const int t = blockIdx.x * blockDim.x + threadIdx.x;
  const int e = t >> 4;
  const int c = t & 15;
  if (e >= E) return;
  const int s = esrc[e];
  const int d = edst[e];
  const float v = evals[e];
  float xv = Xin[(size_t)s * HIDDEN + c];
  if (doRelu) xv = fmaxf(xv, 0.0f);
  atomicAdd(&Z[(size_t)d * HIDDEN + c], v * xv);
}

// ---------------------------------------------------------------------------
// out[N,40] = log_softmax(Z2[N,16] @ W2[16,40]). One wave per 16-row tile.
// 3 column tiles (cols 0-15, 16-31, 32-47 with 40..47 zero-padded),
// K=16 -> 4 WMMA steps per tile = 12 WMMAs/wave. Log-softmax is fused on the
// accumulator layout with 16-lane xor-shuffle reductions.
// ---------------------------------------------------------------------------
__global__ __launch_bounds__(32) void gcn_gemm2_lsm_wmma(
    const float* __restrict__ Z2, const float* __restrict__ W2,
    float* __restrict__ out, int N) {
  const int lane = (int)threadIdx.x;
  const int r0   = blockIdx.x * 16;
  const int col  = lane & 15;
  const int kh   = lane >> 4;
  const int rowA = min(r0 + col, N - 1);
  const float* __restrict__ Zr = Z2 + (size_t)rowA * HIDDEN;

  v8f acc0 = {}, acc1 = {}, acc2 = {};
#pragma unroll
  for (int k0 = 0; k0 < HIDDEN; k0 += 4) {     // 4 steps
    const int ka = k0 + 2 * kh;
    v2f a;
    a.x = Zr[ka];
    a.y = Zr[ka + 1];

    v2f b0;                                    // cols 0..15
    b0.x = W2[ka * NCLS + col];
    b0.y = W2[(ka + 1) * NCLS + col];
    acc0 = wmma_f32_16x16x4(a, b0, acc0);

    v2f b1;                                    // cols 16..31
    b1.x = W2[ka * NCLS + 16 + col];
    b1.y = W2[(ka + 1) * NCLS + 16 + col];
    acc1 = wmma_f32_16x16x4(a, b1, acc1);

    const int c2 = 32 + col;                   // cols 32..47 (pad >= 40)
    v2f b2;
    b2.x = (c2 < NCLS) ? W2[ka * NCLS + c2] : 0.0f;
    b2.y = (c2 < NCLS) ? W2[(ka + 1) * NCLS + c2] : 0.0f;
    acc2 = wmma_f32_16x16x4(a, b2, acc2);
  }

  const bool t2ok = (col < 8);                 // tile2 valid cols: 32..39
  const float ninf = -__builtin_inff();
  const bool fullTile = (r0 + 16 <= N);        // uniform across the wave
  float* __restrict__ Or = out + (size_t)(r0 + 8 * kh) * NCLS + col;

#pragma unroll
  for (int g = 0; g < 8; ++g) {
    const float v0 = acc0[g];
    const float v1 = acc1[g];
    const float v2v = acc2[g];

    float m = fmaxf(fmaxf(v0, v1), t2ok ? v2v : ninf);
#pragma unroll
    for (int off = 8; off >= 1; off >>= 1) m = fmaxf(m, __shfl_xor(m, off, 16));

    float s = expf(v0 - m) + expf(v1 - m) + (t2ok ? expf(v2v - m) : 0.0f);
#pragma unroll
    for (int off = 8; off >= 1; off >>= 1) s += __shfl_xor(s, off, 16);

    const float lse = m + logf(s);

    if (fullTile) {                            // scalar branch, no exec churn
      Or[g * NCLS] = v0 - lse;
      Or[g * NCLS + 16] = v1 - lse;
      if (t2ok) Or[g * NCLS + 32] = v2v - lse;
    } else {
      const int row = r0 + g + 8 * kh;
      if (row < N) {
        out[(size_t)row * NCLS + col] = v0 - lse;
        out[(size_t)row * NCLS + 16 + col] = v1 - lse;
        if (t2ok) out[(size_t)row * NCLS + 32 + col] = v2v - lse;
      }
    }
  }
}

// ---------------------------------------------------------------------------
// Host-side launcher. Inputs (setup_inputs order):
//   0: features [N,100] f32   1: edge_src [E] i32   2: edge_dst [E] i32
//   3: edge_vals [E] f32      4: W1 [100,16] f32    5: W2 [16,40] f32
// Output: log_softmax logits [N,40] f32.
// Uses associativity: relu(spmm(X) @ W1) == relu(spmm(X @ W1)) so the SpMMs
// run on 16-dim vectors instead of 100-dim (6x less random traffic).
// ---------------------------------------------------------------------------
extern "C" void kernel_launch(void* const* d_in, const int* in_sizes, int n_in,
                              void* d_out, int out_size, void* d_ws, size_t ws_size,
                              hipStream_t stream) {
  const float* features = (const float*)d_in[0];
  const int*   esrc     = (const int*)d_in[1];
  const int*   edst     = (const int*)d_in[2];
  const float* evals    = (const float*)d_in[3];
  const float* W1       = (const float*)d_in[4];
  const float* W2       = (const float*)d_in[5];
  float*       out      = (float*)d_out;

  const int N = in_sizes[0] / IN_DIM;   // 50000
  const int E = in_sizes[1];            // 800000

  // Workspace: Y [N,16] | Z1 [N,16] | Z2 [N,16]  (f32)
  float* Y  = (float*)d_ws;
  float* Z1 = Y + (size_t)N * HIDDEN;
  float* Z2 = Z1 + (size_t)N * HIDDEN;

  // 1) Zero the atomic accumulators (Z1, Z2 are contiguous).
  {
    const int n = 2 * N * HIDDEN;
    const int blocks = (n + 255) / 256;
    gcn_zero_kernel<<<blocks, 256, 0, stream>>>(Z1, n);
  }

  // 2) Y = X @ W1  (dense, WMMA f32 16x16x4)
  gcn_gemm1_wmma<<<(N + 15) / 16, 32, 0, stream>>>(features, W1, Y, N);

  // 3) Z1 = A @ Y  (edge-list SpMM, 16-dim messages)
  {
    const int threads = E * 16;
    gcn_spmm_kernel<<<(threads + 255) / 256, 256, 0, stream>>>(
        esrc, edst, evals, Y, Z1, E, /*doRelu=*/0);
  }

  // 4) Z2 = A @ relu(Z1)  (ReLU fused into the gather)
  {
    const int threads = E * 16;
    gcn_spmm_kernel<<<(threads + 255) / 256, 256, 0, stream>>>(
        esrc, edst, evals, Z1, Z2, E, /*doRelu=*/1);
  }

  // 5) out = log_softmax(Z2 @ W2)  (WMMA + fused softmax reduction)
  gcn_gemm2_lsm_wmma<<<(N + 15) / 16, 32, 0, stream>>>(Z2, W2, out, N);
}